// SNN_CV_Layer_2405181686120
// MI455X (gfx1250) — compile-verified
//
#include <hip/hip_runtime.h>
#include <hip/hip_bf16.h>

typedef __attribute__((ext_vector_type(2))) float        v2f;
typedef __attribute__((ext_vector_type(8))) float        v8f;
typedef __attribute__((ext_vector_type(4))) unsigned int v4u;
typedef __attribute__((ext_vector_type(8))) int          v8i;
typedef __attribute__((ext_vector_type(4))) int          v4i;

#define MAXT 100000.0f
#define KTOT 144   // 3*3*16
#define COUT 32

// One wave32 per output pixel. Block = 256 threads = 8 waves. Grid = 1024.
__global__ __launch_bounds__(256) void snn_conv_spike_kernel(
    const float* __restrict__ x,       // [8,32,32,16]
    const float* __restrict__ weight,  // [144,32]
    float* __restrict__ out)           // [8,32,32,32]
{
  __shared__ float          lw[KTOT * COUT];   // 18 KB staged weights (TDM destination)
  __shared__ unsigned       vbits[8][KTOT];    // spike-time bit patterns (sortable as u32)
  __shared__ float          svs[8][KTOT];      // sorted spike times
  __shared__ unsigned short sidx[8][KTOT];     // sorted original indices (weight rows)

  const int tid  = threadIdx.x;
  const int lane = tid & 31;
  const int wv   = tid >> 5;

  // ---- stage weights into LDS via the Tensor Data Mover (wave 0 issues one DMA) ----
  // Degenerate 2D descriptor: 4608 x 1 elements of 4 bytes, tile == tensor.
  if (wv == 0) {
    v4u g0;
    g0.x = 1u;                                   // count=1, is_restore=0, gather off
    g0.y = (unsigned)(uintptr_t)&lw[0];          // lds_addr (LDS byte offset)
    unsigned long long ga = (unsigned long long)(uintptr_t)weight;
    g0.z = (unsigned)(ga & 0xFFFFFFFFull);       // global_addr[31:0]
    g0.w = (unsigned)((ga >> 32) & 0x01FFFFFFull) | 0x80000000u;  // addr[56:32] | type=2

    v8i g1;
    g1[0] = 0x00020000;                          // wg_mask=0, data_size=2 (4 bytes)
    g1[1] = (int)((KTOT * COUT & 0xFFFF) << 16); // tensor_dim0[15:0] = 4608
    g1[2] = (int)(1u << 16);                     // tensor_dim0 hi = 0, tensor_dim1 = 1
    g1[3] = (int)((unsigned)(KTOT * COUT) << 16);// tile_dim0 = 4608
    g1[4] = 1;                                   // tile_dim1 = 1, tile_dim2 = 0
    g1[5] = KTOT * COUT;                         // tensor_dim0_stride = 4608
    g1[6] = 0;                                   // stride hi, tensor_dim1_stride lo
    g1[7] = 0;                                   // tensor_dim1_stride hi

    v4i z4 = {0, 0, 0, 0};                       // groups 2/3: unused dims
    v8i z8 = {0, 0, 0, 0, 0, 0, 0, 0};           // aux operand (6-arg toolchain form)
    __builtin_amdgcn_tensor_load_to_lds(g0, g1, z4, z4, z8, 0);
  }

  const int p  = blockIdx.x * 8 + wv;   // pixel id 0..8191
  const int b  = p >> 10;
  const int hw = p & 1023;
  const int h  = hw >> 5;
  const int w  = hw & 31;

  // ---- phase 1: gather patch, threshold, publish bit patterns ----
  // element t = (kh*3+kw)*16 + c ; lane owns t = lane, lane+32, ... (lanes<16 get a 5th)
  float    myv[5];
  unsigned mybits[5];
  int      myt[5];
  int      myrank[5] = {0, 0, 0, 0, 0};
  const int ne = (lane < 16) ? 5 : 4;
#pragma unroll
  for (int e = 0; e < 5; ++e) {
    int t = lane + 32 * e;
    float v;
    if (e < ne) {
      int c   = t & 15;
      int kwh = t >> 4;
      int kh  = kwh / 3;
      int kw  = kwh - 3 * kh;
      int ih  = h + kh - 1;
      int iw  = w + kw - 1;
      float val = 0.0f;  // SAME zero padding: 0 < 0.9 -> MAX, matches reference
      if ((unsigned)ih < 32u && (unsigned)iw < 32u)
        val = x[(((b * 32 + ih) * 32) + iw) * 16 + c];
      v = (val < 0.9f) ? MAXT : val;
      vbits[wv][t] = __float_as_uint(v);
    } else {
      v = MAXT;
      t = 100000;  // never wins a tiebreak; rank unused
    }
    myv[e]    = v;
    mybits[e] = __float_as_uint(v);
    myt[e]    = t;
  }
  __syncthreads();

  // ---- phase 2: stable rank sort (positive floats: u32 bits order == value order) ----
  for (int j = 0; j < KTOT; ++j) {
    unsigned bj = vbits[wv][j];  // broadcast read
#pragma unroll
    for (int e = 0; e < 5; ++e)
      myrank[e] += (int)((bj < mybits[e]) | ((bj == mybits[e]) & (j < myt[e])));
  }
#pragma unroll
  for (int e = 0; e < 5; ++e) {
    if (e < ne) {
      svs[wv][myrank[e]]  = myv[e];
      sidx[wv][myrank[e]] = (unsigned short)myt[e];
    }
  }
  // weights must be resident before any wave reads lw[]; the TDM-issuing wave
  // drains TENSORcnt (no-op for the others), then the block barrier publishes LDS.
  __builtin_amdgcn_s_wait_tensorcnt(0);
  __syncthreads();

  // ---- phase 3: WMMA triangular prefix scan over 9 chunks of 16 positions ----
  const int n16 = lane & 15;  // A-matrix row M; also channel within half
  const int grp = lane >> 4;  // lane half
  const int m   = n16;

  for (int half = 0; half < 2; ++half) {
    const int ch = n16 + 16 * half;  // output channel handled by this lane
    float cs = 0.0f, ci = 0.0f;      // per-channel carries (wsum / wisum)
    float minv = MAXT;

    for (int c0 = 0; c0 < 9; ++c0) {
      const int base = c0 * 16;

      // Operand build. B (4x16) element = lanehalf*2 + vgpr -> rows {2g, 2g+1}.
      v2f B[4], A[4], Av[4];
#pragma unroll
      for (int k = 0; k < 4; ++k) {
        int jx = 4 * k + 2 * grp;  // column (for A) / row (for B) of .x ; .y is jx+1
        int r0 = (int)sidx[wv][base + jx];
        int r1 = (int)sidx[wv][base + jx + 1];
        v2f bb; bb.x = lw[r0 * COUT + ch]; bb.y = lw[r1 * COUT + ch];
        B[k] = bb;
        float vx = svs[wv][base + jx];
        float vy = svs[wv][base + jx + 1];
        v2f aa; aa.x = (m >= jx) ? 1.0f : 0.0f; aa.y = (m >= jx + 1) ? 1.0f : 0.0f;
        A[k] = aa;
        v2f av; av.x = (m >= jx) ? vx : 0.0f;   av.y = (m >= jx + 1) ? vy : 0.0f;
        Av[k] = av;
      }

      v8f Ds = {cs, cs, cs, cs, cs, cs, cs, cs};
      v8f Di = {ci, ci, ci, ci, ci, ci, ci, ci};
      // wsum prefix: D = T(16x16 lower-tri ones) @ Wp + carry, K=16 as 4 chained K=4 WMMAs
#pragma unroll
      for (int k = 0; k < 4; ++k)
        Ds = __builtin_amdgcn_wmma_f32_16x16x4_f32(false, A[k], false, B[k],
                                                   (short)0, Ds, false, false);
      // wisum prefix: D = Tv @ Wp + carry (Tv columns scaled by sorted spike times)
#pragma unroll
      for (int k = 0; k < 4; ++k)
        Di = __builtin_amdgcn_wmma_f32_16x16x4_f32(false, Av[k], false, B[k],
                                                   (short)0, Di, false, false);

      // Post-process my rows: M = 8*grp + g (D layout: lane<16 -> M=g, lane>=16 -> M=g+8)
#pragma unroll
      for (int g = 0; g < 8; ++g) {
        int   kpos  = base + 8 * grp + g;
        float wsum  = Ds[g];
        float wisum = Di[g];
        float denom = fminf(fmaxf(wsum - 1.0f, 1e-10f), 1e10f);
        float o     = wisum / denom;
        float vsk   = svs[wv][kpos];
        float vlf   = (kpos == KTOT - 1) ? MAXT : svs[wv][kpos + 1];
        o = (wsum < 1.0f) ? MAXT : o;
        o = (o < vsk)     ? MAXT : o;
        o = (o > vlf)     ? MAXT : o;
        minv = fminf(minv, o);
      }

      // carry = D row 15 (lives in lane 16+n16, element 7) broadcast to the lane pair
      cs = __shfl(Ds[7], 16 + n16, 32);
      ci = __shfl(Di[7], 16 + n16, 32);
    }

    // combine mins across the lane pair covering this channel, then store
    float other = __shfl(minv, lane ^ 16, 32);
    minv = fminf(minv, other);
    if (lane < 16) out[p * COUT + ch] = minv;
  }
}

extern "C" void kernel_launch(void* const* d_in, const int* in_sizes, int n_in,
                              void* d_out, int out_size, void* d_ws, size_t ws_size,
                              hipStream_t stream) {
  const float* x      = (const float*)d_in[0];
  const float* weight = (const float*)d_in[1];
  float*       out    = (float*)d_out;
  // 8192 pixels, one wave32 each, 8 waves per block
  snn_conv_spike_kernel<<<dim3(1024), dim3(256), 0, stream>>>(x, weight, out);
}